// NormalsLoss_71322226917416
// MI455X (gfx1250) — compile-verified
//
#include <hip/hip_runtime.h>
#include <math.h>

typedef float v2f __attribute__((ext_vector_type(2)));
typedef float v4f __attribute__((ext_vector_type(4)));
typedef float v8f __attribute__((ext_vector_type(8)));

#define JCHUNKS 4
#define STAGE_PTS 2048                 // data points staged in LDS per stage (32 KB)
#define STAGE_TILES (STAGE_PTS / 16)   // 128 j-tiles per stage (power of two)
#define PIPE 4                         // software-pipeline depth (tiles in flight)

// ---------------------------------------------------------------------------
// Kernel 1: pack rec -> [-2x,-2y,-2z,1]  and  data -> [x,y,z,|d|^2]
// so each WMMA operand lane loads exactly its two K-dwords with one b64 load.
// ---------------------------------------------------------------------------
__global__ void nl_pack(const float* __restrict__ rec,
                        const float* __restrict__ data,
                        v4f* __restrict__ recP,
                        v4f* __restrict__ datP,
                        int N, int M) {
    int i = blockIdx.x * blockDim.x + threadIdx.x;
    if (i < N) {
        float x = rec[3 * i], y = rec[3 * i + 1], z = rec[3 * i + 2];
        v4f r; r.x = -2.0f * x; r.y = -2.0f * y; r.z = -2.0f * z; r.w = 1.0f;
        recP[i] = r;
    }
    if (i < M) {
        float x = data[3 * i], y = data[3 * i + 1], z = data[3 * i + 2];
        v4f d; d.x = x; d.y = y; d.z = z; d.w = x * x + y * y + z * z;
        datP[i] = d;
    }
}

// ---------------------------------------------------------------------------
// Kernel 2: WMMA argmin.
//  - Block (8 waves) owns one j-chunk; wave w owns i-block (blockGroup*8+w).
//  - Chunk data staged through 32 KB LDS tiles shared by all 8 waves.
//  - B operands are kept in a PIPE-deep register buffer; the next group of
//    PIPE tiles is prefetched (ds_load_b64) before the current group is
//    consumed, so no WMMA waits on a same-group load.
//  - Keys are d^2 - 2*rec.data from v_wmma_f32_16x16x4_f32 (r^2 row term is
//    constant under argmin and dropped).
// ---------------------------------------------------------------------------
__global__ void __launch_bounds__(256) nl_argmin(const v4f* __restrict__ recP,
                                                 const v4f* __restrict__ datP,
                                                 float* __restrict__ pval,
                                                 int*   __restrict__ pidx,
                                                 int N, int M) {
    const int lane = threadIdx.x & 31;
    const int w    = threadIdx.x >> 5;                 // wave in block (0..7)
    const int ibGroup = blockIdx.x / JCHUNKS;
    const int chunk   = blockIdx.x % JCHUNKS;
    int i0 = (ibGroup * 8 + w) << 4;
    const bool active = (i0 < N);
    if (!active) i0 = 0;                               // keep loads in-bounds

    const int Mc        = M / JCHUNKS;                 // points per chunk
    const int chunkBase = chunk * Mc;
    const int nStages   = Mc / STAGE_PTS;

    __shared__ v4f sdat[STAGE_PTS];

    // A operand: lane l<16 holds row (i0+l), K=0,1 ; lane l>=16 holds K=2,3.
    const float* aptr = (const float*)(recP + i0 + (lane & 15)) + ((lane >> 4) << 1);
    v2f a = *(const v2f*)aptr;

    float minv[8];
    int   minj[8];
#pragma unroll
    for (int v = 0; v < 8; ++v) { minv[v] = 3.4028235e38f; minj[v] = 0; }

    // per-lane fixed float offset inside a 16-point LDS tile (64 floats)
    const int loff = ((lane & 15) << 2) + ((lane >> 4) << 1);
    const float* lbase = (const float*)sdat;

    for (int st = 0; st < nStages; ++st) {
        const int base = chunkBase + st * STAGE_PTS;   // global point index
        __syncthreads();
#pragma unroll
        for (int k = 0; k < STAGE_PTS / 256; ++k) {
            int idx = threadIdx.x + k * 256;
            sdat[idx] = datP[base + idx];
        }
        __syncthreads();

        // prime the pipeline: first PIPE tiles' B operands
        v2f b[PIPE];
#pragma unroll
        for (int k = 0; k < PIPE; ++k)
            b[k] = *(const v2f*)(lbase + (k << 6) + loff);

        for (int t = 0; t < STAGE_TILES; t += PIPE) {
            // prefetch the next group (wraps in-bounds on the last group;
            // those values are never consumed)
            v2f bn[PIPE];
#pragma unroll
            for (int k = 0; k < PIPE; ++k) {
                int tn = (t + PIPE + k) & (STAGE_TILES - 1);
                bn[k] = *(const v2f*)(lbase + (tn << 6) + loff);
            }

#pragma unroll
            for (int k = 0; k < PIPE; ++k) {
                v8f c = {0.f, 0.f, 0.f, 0.f, 0.f, 0.f, 0.f, 0.f};
                v8f d = __builtin_amdgcn_wmma_f32_16x16x4_f32(
                    false, a, false, b[k], (short)0, c, false, false);

                const int j0 = base + ((t + k) << 4);
#pragma unroll
                for (int v = 0; v < 8; ++v) {
                    float dv = d[v];
                    bool p = dv < minv[v];
                    minv[v] = p ? dv : minv[v];
                    minj[v] = p ? j0 : minj[v];
                }
            }

#pragma unroll
            for (int k = 0; k < PIPE; ++k) b[k] = bn[k];
        }
    }

    // Complete the index: this lane always scanned column offset (lane&15).
#pragma unroll
    for (int v = 0; v < 8; ++v) minj[v] += (lane & 15);

    // Butterfly min-reduce within each 16-lane half (masks < 16 stay in-half).
    // Tie-break: smaller index wins, matching jnp.argmin (first minimum).
#pragma unroll
    for (int m = 1; m <= 8; m <<= 1) {
#pragma unroll
        for (int v = 0; v < 8; ++v) {
            float ov = __shfl_xor(minv[v], m, 32);
            int   oi = __shfl_xor(minj[v], m, 32);
            if (ov < minv[v] || (ov == minv[v] && oi < minj[v])) {
                minv[v] = ov; minj[v] = oi;
            }
        }
    }

    // Lane 0 holds rows i0..i0+7 (VGPR v -> row i0+v); lane 16 rows i0+8..15.
    if (active && (lane & 15) == 0) {
        int rbase = i0 + ((lane >> 4) << 3);
#pragma unroll
        for (int v = 0; v < 8; ++v) {
            int i = rbase + v;
            pval[(i << 2) | chunk] = minv[v];
            pidx[(i << 2) | chunk] = minj[v];
        }
    }
}

// ---------------------------------------------------------------------------
// Kernel 3: combine the JCHUNKS partial argmins per row, gather data_normals,
// compute ||dn - rn||, block-reduce to one partial sum per block.
// ---------------------------------------------------------------------------
__global__ void nl_finalize(const float* __restrict__ pval,
                            const int*   __restrict__ pidx,
                            const float* __restrict__ recN,
                            const float* __restrict__ datN,
                            float* __restrict__ blockSums,
                            int N) {
    int i = blockIdx.x * blockDim.x + threadIdx.x;
    float s = 0.0f;
    if (i < N) {
        float bv = pval[i << 2];
        int   bi = pidx[i << 2];
#pragma unroll
        for (int c = 1; c < JCHUNKS; ++c) {
            float v = pval[(i << 2) | c];
            int   j = pidx[(i << 2) | c];
            if (v < bv || (v == bv && j < bi)) { bv = v; bi = j; }
        }
        float dx = datN[3 * bi]     - recN[3 * i];
        float dy = datN[3 * bi + 1] - recN[3 * i + 1];
        float dz = datN[3 * bi + 2] - recN[3 * i + 2];
        s = sqrtf(dx * dx + dy * dy + dz * dz);
    }
#pragma unroll
    for (int m = 16; m >= 1; m >>= 1) s += __shfl_xor(s, m, 32);
    __shared__ float wsum[8];
    int lane = threadIdx.x & 31;
    int wv   = threadIdx.x >> 5;
    if (lane == 0) wsum[wv] = s;
    __syncthreads();
    if (wv == 0) {
        int nw = blockDim.x >> 5;
        float t = (lane < nw) ? wsum[lane] : 0.0f;
#pragma unroll
        for (int m = 4; m >= 1; m >>= 1) t += __shfl_xor(t, m, 32);
        if (lane == 0) blockSums[blockIdx.x] = t;
    }
}

// ---------------------------------------------------------------------------
// Kernel 4: deterministic final sum of block partials -> mean.
// ---------------------------------------------------------------------------
__global__ void nl_sum(const float* __restrict__ blockSums, int nBlocks,
                       float* __restrict__ out, float invN) {
    float s = 0.0f;
    for (int i = threadIdx.x; i < nBlocks; i += 32) s += blockSums[i];
#pragma unroll
    for (int m = 16; m >= 1; m >>= 1) s += __shfl_xor(s, m, 32);
    if (threadIdx.x == 0) out[0] = s * invN;
}

// ---------------------------------------------------------------------------
extern "C" void kernel_launch(void* const* d_in, const int* in_sizes, int n_in,
                              void* d_out, int out_size, void* d_ws, size_t ws_size,
                              hipStream_t stream) {
    (void)n_in; (void)out_size; (void)ws_size;
    const float* rec  = (const float*)d_in[0];
    const float* data = (const float*)d_in[1];
    const float* recN = (const float*)d_in[2];
    const float* datN = (const float*)d_in[3];
    float* out = (float*)d_out;

    const int N = in_sizes[0] / 3;   // 16384
    const int M = in_sizes[1] / 3;   // 16384

    // workspace layout
    v4f*   recP      = (v4f*)d_ws;                         // N * 16 B
    v4f*   datP      = recP + N;                           // M * 16 B
    float* pval      = (float*)(datP + M);                 // N * JCHUNKS * 4 B
    int*   pidx      = (int*)(pval + (size_t)N * JCHUNKS); // N * JCHUNKS * 4 B
    float* blockSums = (float*)(pidx + (size_t)N * JCHUNKS);

    // 1) pack operands
    int mx = (N > M) ? N : M;
    nl_pack<<<(mx + 255) / 256, 256, 0, stream>>>(rec, data, recP, datP, N, M);

    // 2) WMMA argmin: blocks = ceil(N/16/8) i-groups * JCHUNKS chunks
    int ibGroups = (N / 16 + 7) / 8;                       // 128
    nl_argmin<<<ibGroups * JCHUNKS, 256, 0, stream>>>(recP, datP, pval, pidx, N, M);

    // 3) combine + normal diff + per-block partial sums
    int finBlocks = (N + 255) / 256;                       // 64
    nl_finalize<<<finBlocks, 256, 0, stream>>>(pval, pidx, recN, datN, blockSums, N);

    // 4) deterministic scalar mean
    nl_sum<<<1, 32, 0, stream>>>(blockSums, finBlocks, out, 1.0f / (float)N);
}